// get_loss_67095979098154
// MI455X (gfx1250) — compile-verified
//
#include <hip/hip_runtime.h>
#include <hip/hip_bf16.h>

typedef __attribute__((ext_vector_type(2))) float v2f;
typedef __attribute__((ext_vector_type(4))) float v4f;
typedef __attribute__((ext_vector_type(8))) float v8f;

#define NPTS 4096
#define NB 4
#define NCLS 13
#define ALPHA_C 10.0f
#define MARGIN0 1.0f
#define MARGIN1 2.0f
#define ROWS_PER_BLOCK 8
#define TPB 256
#define BLOCKS_PER_BATCH (NPTS / ROWS_PER_BLOCK)   // 512
#define NBLK_MAIN (NB * BLOCKS_PER_BATCH)          // 2048

// ---------------------------------------------------------------------------
// Sum of a value across all 32 lanes of a wave using V_WMMA_F32_16X16X4_F32.
// v in A VGPR0, 0 in VGPR1, B = ones => D[m][n] = v[m] + v[m+16].
// Per-lane sum of the 8 D VGPRs gives half the total; one shfl_xor(16)
// completes it (result valid in every lane). Callers guarantee EXEC all-ones.
// ---------------------------------------------------------------------------
__device__ __forceinline__ float wave_sum32_wmma(float v) {
  v2f a; a[0] = v;    a[1] = 0.0f;
  v2f o; o[0] = 1.0f; o[1] = 1.0f;
  v8f c = {};
  v8f d = __builtin_amdgcn_wmma_f32_16x16x4_f32(
      /*neg_a=*/false, a, /*neg_b=*/false, o,
      /*c_mod=*/(short)0, c, /*reuse_a=*/false, /*reuse_b=*/false);
  float s = d[0] + d[1] + d[2] + d[3] + d[4] + d[5] + d[6] + d[7];
  s += __shfl_xor(s, 16, 32);
  return s;
}

// All 256 threads must call (EXEC full for every wave -> WMMA is legal).
__device__ __forceinline__ float block_sum256_wmma(float v, float* wsum) {
  const float s = wave_sum32_wmma(v);
  const int wave = threadIdx.x >> 5;
  const int lane = threadIdx.x & 31;
  if (lane == 0) wsum[wave] = s;
  __syncthreads();
  float tot = 0.0f;
#pragma unroll
  for (int w = 0; w < TPB / 32; ++w) tot += wsum[w];
  __syncthreads();  // allow wsum reuse by the caller
  return tot;
}

// ---------------------------------------------------------------------------
// Main fused pass: one wave per row of pred_simmat. Uniform-trip-count loop
// (scalar branch only), 2x unrolled non-temporal float4 stream (268MB read
// once; > 192MB L2, keep it out). Labels packed 1 byte/column in 4KB of LDS.
// intersection/union are wave-uniform ballot popcounts on the SALU; loss uses
// two independent VALU accumulators, reduced at the end via the matrix pipe.
// ---------------------------------------------------------------------------
__global__ __launch_bounds__(TPB) void sim_cf_kernel(
    const float* __restrict__ simmat, const float* __restrict__ cfmat,
    const int* __restrict__ label, float* __restrict__ simPartial,
    float* __restrict__ sqPartial) {
  __shared__ unsigned int labp[NPTS / 4];  // (class<<4)|inst, 4 cols per uint
  __shared__ float wLoss[ROWS_PER_BLOCK];
  __shared__ float wSq[ROWS_PER_BLOCK];

  const int b    = blockIdx.x / BLOCKS_PER_BATCH;
  const int row0 = (blockIdx.x % BLOCKS_PER_BATCH) * ROWS_PER_BLOCK;

  const int* __restrict__ cls = label + (size_t)b * 2 * NPTS;  // label[b,0,:]
  const int* __restrict__ ins = cls + NPTS;                    // label[b,1,:]

  for (int q = threadIdx.x; q < NPTS / 4; q += TPB) {
    unsigned int p = 0;
#pragma unroll
    for (int k = 0; k < 4; ++k) {
      const int j = q * 4 + k;
      p |= (((((unsigned)cls[j]) & 0xFu) << 4) | (((unsigned)ins[j]) & 0xFu))
           << (8 * k);
    }
    labp[q] = p;
  }
  __syncthreads();

  const int wave = threadIdx.x >> 5;
  const int lane = threadIdx.x & 31;
  const int i = row0 + wave;
  const unsigned my    = (labp[i >> 2] >> (8 * (i & 3))) & 0xFFu;
  const unsigned myCls = my >> 4;
  const unsigned myIns = my & 0xFu;

  const float* __restrict__ rowp =
      simmat + ((size_t)b * NPTS + i) * (size_t)NPTS;

  float loss0 = 0.0f, loss1 = 0.0f;
  unsigned interC = 0u;
  unsigned uniC = 0u;
  const int laneOff = lane * 4;

#pragma unroll 1
  for (int it = 0; it < NPTS / 256; ++it) {  // uniform trip count: scalar loop
    const int base = it * 256;
    __builtin_prefetch(rowp + base + 4096, 0, 0);  // 16KB ahead, speculative
    const v4f sA =
        __builtin_nontemporal_load((const v4f*)(rowp + base + laneOff));
    const v4f sB =
        __builtin_nontemporal_load((const v4f*)(rowp + base + 128 + laneOff));
    const unsigned pkA = labp[(base + laneOff) >> 2];
    const unsigned pkB = labp[(base + 128 + laneOff) >> 2];
#pragma unroll
    for (int k = 0; k < 4; ++k) {
      {
        const float s = sA[k];
        const unsigned pb = (pkA >> (8 * k)) & 0xFFu;
        const bool g  = ((pb & 0xFu) == myIns);
        const bool sm = ((pb >> 4) == myCls);
        const bool pg = (s < MARGIN0);
        const float relu1 = fmaxf(MARGIN0 - s, 0.0f);
        const float relu2 = fmaxf(MARGIN1 - s, 0.0f);
        const float lns = sm ? (ALPHA_C * relu1) : relu2;
        loss0 += g ? s : lns;
        const unsigned bg = __builtin_amdgcn_ballot_w32(g);
        const unsigned bp = __builtin_amdgcn_ballot_w32(pg);
        interC += (unsigned)__builtin_popcount(bg & bp);
        uniC   += (unsigned)__builtin_popcount(bg | bp);
      }
      {
        const float s = sB[k];
        const unsigned pb = (pkB >> (8 * k)) & 0xFFu;
        const bool g  = ((pb & 0xFu) == myIns);
        const bool sm = ((pb >> 4) == myCls);
        const bool pg = (s < MARGIN0);
        const float relu1 = fmaxf(MARGIN0 - s, 0.0f);
        const float relu2 = fmaxf(MARGIN1 - s, 0.0f);
        const float lns = sm ? (ALPHA_C * relu1) : relu2;
        loss1 += g ? s : lns;
        const unsigned bg = __builtin_amdgcn_ballot_w32(g);
        const unsigned bp = __builtin_amdgcn_ballot_w32(pg);
        interC += (unsigned)__builtin_popcount(bg & bp);
        uniC   += (unsigned)__builtin_popcount(bg | bp);
      }
    }
  }

  // Loss reduction via the matrix pipe (EXEC all-ones here).
  const float lossSum = wave_sum32_wmma(loss0 + loss1);

  if (lane == 0) {
    wLoss[wave] = lossSum;
    const float cf = (float)interC / (float)uniC;  // uni >= 1 (diagonal)
    const float dd = cf - cfmat[(size_t)b * NPTS + i];
    wSq[wave] = dd * dd;
  }
  __syncthreads();
  if (threadIdx.x == 0) {
    float ls = 0.0f, sq = 0.0f;
#pragma unroll
    for (int w = 0; w < ROWS_PER_BLOCK; ++w) { ls += wLoss[w]; sq += wSq[w]; }
    simPartial[blockIdx.x] = ls;
    sqPartial[blockIdx.x]  = sq;  // grouped by batch: [b*512 + blkInBatch]
  }
}

// ---------------------------------------------------------------------------
// sem_loss gather: one thread per (b, n).
// ---------------------------------------------------------------------------
__global__ __launch_bounds__(TPB) void sem_kernel(
    const float* __restrict__ semmat, const int* __restrict__ label,
    float* __restrict__ semPartial) {
  __shared__ float wsum[TPB / 32];
  const int idx = blockIdx.x * TPB + threadIdx.x;  // grid covers exactly B*N
  const int b = idx / NPTS;
  const int n = idx % NPTS;
  const int c = label[(size_t)b * 2 * NPTS + n];
  const float g = semmat[((size_t)b * NPTS + n) * NCLS + c];
  const float tot = block_sum256_wmma(g, wsum);
  if (threadIdx.x == 0) semPartial[blockIdx.x] = tot;
}

// ---------------------------------------------------------------------------
// Final reduction: one block. Deterministic (no float atomics anywhere).
// ---------------------------------------------------------------------------
__global__ __launch_bounds__(TPB) void finalize_kernel(
    const float* __restrict__ simPartial, const float* __restrict__ sqPartial,
    const float* __restrict__ semPartial, float* __restrict__ out) {
  __shared__ float wsum[TPB / 32];
  const int t = threadIdx.x;

  float s = 0.0f;
  for (int k = t; k < NBLK_MAIN; k += TPB) s += simPartial[k];
  const float simTot = block_sum256_wmma(s, wsum);

  float sqTot[NB];
#pragma unroll
  for (int bb = 0; bb < NB; ++bb) {
    float p = 0.0f;
    for (int k = t; k < BLOCKS_PER_BATCH; k += TPB)
      p += sqPartial[bb * BLOCKS_PER_BATCH + k];
    sqTot[bb] = block_sum256_wmma(p, wsum);
  }

  const int nSemBlk = (NB * NPTS) / TPB;  // 64
  float sp = (t < nSemBlk) ? semPartial[t] : 0.0f;
  const float semTot = block_sum256_wmma(sp, wsum);

  if (t == 0) {
    out[0] = simTot / ((float)NB * (float)NPTS * (float)NPTS);
    float cf = 0.0f;
#pragma unroll
    for (int bb = 0; bb < NB; ++bb) cf += sqrtf(sqTot[bb]);
    out[1] = cf / (float)NB;
    out[2] = -semTot / ((float)NB * (float)NPTS);
  }
}

extern "C" void kernel_launch(void* const* d_in, const int* in_sizes, int n_in,
                              void* d_out, int out_size, void* d_ws,
                              size_t ws_size, hipStream_t stream) {
  (void)in_sizes; (void)n_in; (void)out_size; (void)ws_size;
  const float* simmat = (const float*)d_in[0];  // (B,N,N) f32
  const float* cfmat  = (const float*)d_in[1];  // (B,N)   f32
  const float* semmat = (const float*)d_in[2];  // (B,N,C) f32
  const int*   label  = (const int*)d_in[3];    // (B,2,N) int32

  float* out = (float*)d_out;
  float* ws = (float*)d_ws;
  float* simPartial = ws;                  // 2048 floats
  float* sqPartial  = ws + NBLK_MAIN;      // 2048 floats
  float* semPartial = ws + 2 * NBLK_MAIN;  // 64 floats

  sim_cf_kernel<<<NBLK_MAIN, TPB, 0, stream>>>(simmat, cfmat, label,
                                               simPartial, sqPartial);
  sem_kernel<<<(NB * NPTS) / TPB, TPB, 0, stream>>>(semmat, label, semPartial);
  finalize_kernel<<<1, TPB, 0, stream>>>(simPartial, sqPartial, semPartial, out);
}